// SMOTELayer_24395414242037
// MI455X (gfx1250) — compile-verified
//
#include <hip/hip_runtime.h>
#include <math.h>
#include <stdint.h>

#define DD 512
#define BS 32768
#define PP 8192
#define NNEW (PP * 3)          // 24576 synthesized rows
#define OUTROWS (BS + NNEW)    // 57344
#define EPSV 1e-5f

typedef __attribute__((ext_vector_type(2))) float v2f;
typedef __attribute__((ext_vector_type(8))) float v8f;

// ---------------------------------------------------------------------------
// GEMM: Y[M x 512] = X[M x 512] * W[512 x 512] + bias, via V_WMMA_F32_16X16X4_F32
// block = 128 threads = 4 waves; each wave computes a 32(M) x 64(N) tile:
// 2 A fragments x 4 B fragments -> 8 accumulators (64 VGPRs of C).
// f32 WMMA layouts (wave32):
//   A 16x4 : lane holds row m=lane%16, ks {2*(lane/16), 2*(lane/16)+1}
//   B 4x16 : lane holds col n=lane%16, same ks
//   C 16x16: vgpr r, lane l -> row r+8*(l/16), col l%16
// ---------------------------------------------------------------------------
__global__ __launch_bounds__(128) void gemm_bias_f32(
    const float* __restrict__ X, const float* __restrict__ W,
    const float* __restrict__ bias, float* __restrict__ Y, int M)
{
    const int lane = threadIdx.x & 31;
    const int wave = threadIdx.x >> 5;
    const int m0 = (blockIdx.x * 4 + wave) * 32;
    const int n0 = blockIdx.y * 64;
    if (m0 >= M) return;

    const int half = lane >> 4;   // 0 or 1 -> which K pair
    const int l16  = lane & 15;

    v8f acc[8] = {};                                 // [t*2 + u], t: N tile, u: M half
    const float* arow0 = X + (size_t)(m0 + l16) * DD;
    const float* arow1 = arow0 + (size_t)16 * DD;

    for (int k = 0; k < DD; k += 4) {
        const int kk = k + 2 * half;
        v2f a0 = *(const v2f*)(arow0 + kk);
        v2f a1 = *(const v2f*)(arow1 + kk);
        const float* base = W + (size_t)kk * DD + n0 + l16;
        v2f b0, b1, b2, b3;
        b0.x = base[0];  b0.y = base[DD];
        b1.x = base[16]; b1.y = base[DD + 16];
        b2.x = base[32]; b2.y = base[DD + 32];
        b3.x = base[48]; b3.y = base[DD + 48];
        acc[0] = __builtin_amdgcn_wmma_f32_16x16x4_f32(false, a0, false, b0, (short)0, acc[0], false, false);
        acc[1] = __builtin_amdgcn_wmma_f32_16x16x4_f32(false, a1, false, b0, (short)0, acc[1], false, false);
        acc[2] = __builtin_amdgcn_wmma_f32_16x16x4_f32(false, a0, false, b1, (short)0, acc[2], false, false);
        acc[3] = __builtin_amdgcn_wmma_f32_16x16x4_f32(false, a1, false, b1, (short)0, acc[3], false, false);
        acc[4] = __builtin_amdgcn_wmma_f32_16x16x4_f32(false, a0, false, b2, (short)0, acc[4], false, false);
        acc[5] = __builtin_amdgcn_wmma_f32_16x16x4_f32(false, a1, false, b2, (short)0, acc[5], false, false);
        acc[6] = __builtin_amdgcn_wmma_f32_16x16x4_f32(false, a0, false, b3, (short)0, acc[6], false, false);
        acc[7] = __builtin_amdgcn_wmma_f32_16x16x4_f32(false, a1, false, b3, (short)0, acc[7], false, false);
    }

    #pragma unroll
    for (int t = 0; t < 4; ++t) {
        const int col = n0 + t * 16 + l16;
        const float bv = bias[col];
        #pragma unroll
        for (int u = 0; u < 2; ++u) {
            #pragma unroll
            for (int r = 0; r < 8; ++r) {
                const int row = m0 + u * 16 + r + 8 * half;
                Y[(size_t)row * DD + col] = acc[t * 2 + u][r] + bv;
            }
        }
    }
}

// ---------------------------------------------------------------------------
// BatchNorm stats: coalesced partial sums + atomic accumulation per column.
// ---------------------------------------------------------------------------
__global__ __launch_bounds__(512) void col_stats_accum(
    const float* __restrict__ Y, float* __restrict__ sum,
    float* __restrict__ sumsq, int M)
{
    const int c = threadIdx.x;
    const int rows = M / gridDim.x;
    const int r0 = blockIdx.x * rows;
    float s = 0.f, q = 0.f;
    for (int r = 0; r < rows; ++r) {
        float v = Y[(size_t)(r0 + r) * DD + c];
        s += v;
        q += v * v;
    }
    atomicAdd(&sum[c], s);
    atomicAdd(&sumsq[c], q);
}

__global__ void col_stats_final(const float* __restrict__ sum,
                                const float* __restrict__ sumsq,
                                float* __restrict__ mean,
                                float* __restrict__ invstd, int M)
{
    const int c = blockIdx.x * blockDim.x + threadIdx.x;
    if (c >= DD) return;
    const float m = sum[c] / (float)M;
    const float v = sumsq[c] / (float)M - m * m;
    mean[c] = m;
    invstd[c] = rsqrtf(v + EPSV);
}

__global__ void zero_f32(float* __restrict__ p, int n)
{
    const int i = blockIdx.x * blockDim.x + threadIdx.x;
    if (i < n) p[i] = 0.f;
}

// normalize + swish (elementwise, safe in-place)
__global__ __launch_bounds__(256) void bn_swish(
    const float* __restrict__ Y, const float* __restrict__ mean,
    const float* __restrict__ invstd, const float* __restrict__ g,
    const float* __restrict__ beta, float* __restrict__ H, int M)
{
    const size_t total = (size_t)M * DD;
    size_t i = (size_t)blockIdx.x * blockDim.x + threadIdx.x;
    const size_t step = (size_t)gridDim.x * blockDim.x;
    for (; i < total; i += step) {
        const int c = (int)(i & (DD - 1));
        const float s = g[c] * (Y[i] - mean[c]) * invstd[c] + beta[c];
        H[i] = s / (1.f + expf(-s));
    }
}

// ---------------------------------------------------------------------------
// Row squared norms for the first PP rows of X. One wave per row.
// ---------------------------------------------------------------------------
__global__ __launch_bounds__(256) void row_sq(const float* __restrict__ X,
                                              float* __restrict__ sq)
{
    const int wave = threadIdx.x >> 5;
    const int lane = threadIdx.x & 31;
    const int row = blockIdx.x * 8 + wave;
    const float* x = X + (size_t)row * DD;
    float s = 0.f;
    for (int k = lane; k < DD; k += 32) {
        float v = x[k];
        s += v * v;
    }
    #pragma unroll
    for (int off = 16; off; off >>= 1) s += __shfl_xor(s, off, 32);
    if (lane == 0) sq[row] = s;
}

// ---------------------------------------------------------------------------
// Fused Gram + top-3: one wave per 16-row i-tile; sweep j in 64-wide tiles.
// The 16x512 A tile (contiguous 32KB of x) is staged once into LDS with
// GLOBAL_LOAD_ASYNC_TO_LDS_B128 (ASYNCcnt), then reused for all 128 j-tiles;
// A fragments come from LDS, B fragments are b64 global loads.
// neg_dist[i][j] = 2*(x_i . x_j) - sq_i - sq_j; 16x68-padded LDS staging of the
// result tile avoids bank conflicts between half-wave row groups; 16 row-owner
// lanes keep a running top-3 (ascending-j strict-> scan == jax top_k ties).
// ---------------------------------------------------------------------------
__global__ __launch_bounds__(32) void knn_top3(
    const float* __restrict__ Xp, const float* __restrict__ sq,
    int* __restrict__ idx)
{
    __shared__ float As[16 * DD];     // 32 KB A tile
    __shared__ float dist[16][68];

    const int lane = threadIdx.x;
    const int m0 = blockIdx.x * 16;
    const int half = lane >> 4;
    const int l16 = lane & 15;

    // ---- async copy of the 32KB A tile (rows m0..m0+15 are contiguous) ----
    {
        const char* src = (const char*)(Xp + (size_t)m0 * DD);
        const uint32_t dst = (uint32_t)(uintptr_t)(void*)As;
        #pragma unroll 4
        for (int it = 0; it < 64; ++it) {
            const uint32_t boff = (uint32_t)(it * 32 + lane) * 16u;
            asm volatile("global_load_async_to_lds_b128 %0, %1, off"
                         :: "v"(dst + boff), "v"(src + boff)
                         : "memory");
        }
    }

    float sqi[8];
    #pragma unroll
    for (int r = 0; r < 8; ++r) sqi[r] = sq[m0 + r + 8 * half];

    float b0v = -1e30f, b1v = -1e30f, b2v = -1e30f;
    int b0i = 0, b1i = 0, b2i = 0;

    asm volatile("s_wait_asynccnt 0x0" ::: "memory");
    __syncthreads();

    const int aoff = l16 * DD;

    for (int n0 = 0; n0 < PP; n0 += 64) {
        v8f c0 = {}, c1 = {}, c2 = {}, c3 = {};
        const float* j0 = Xp + (size_t)(n0 + l16) * DD;   // tile t cols at +16*t*DD
        for (int k = 0; k < DD; k += 4) {
            const int kk = k + 2 * half;
            v2f a = *(const v2f*)&As[aoff + kk];          // ds_load_b64
            v2f b0 = *(const v2f*)(j0 + kk);
            v2f b1 = *(const v2f*)(j0 + 16 * DD + kk);
            v2f b2 = *(const v2f*)(j0 + 32 * DD + kk);
            v2f b3 = *(const v2f*)(j0 + 48 * DD + kk);
            c0 = __builtin_amdgcn_wmma_f32_16x16x4_f32(false, a, false, b0, (short)0, c0, false, false);
            c1 = __builtin_amdgcn_wmma_f32_16x16x4_f32(false, a, false, b1, (short)0, c1, false, false);
            c2 = __builtin_amdgcn_wmma_f32_16x16x4_f32(false, a, false, b2, (short)0, c2, false, false);
            c3 = __builtin_amdgcn_wmma_f32_16x16x4_f32(false, a, false, b3, (short)0, c3, false, false);
        }
        v8f cacc[4] = {c0, c1, c2, c3};
        #pragma unroll
        for (int t = 0; t < 4; ++t) {
            const float sqj = sq[n0 + t * 16 + l16];
            #pragma unroll
            for (int r = 0; r < 8; ++r) {
                dist[r + 8 * half][t * 16 + l16] = 2.f * cacc[t][r] - sqi[r] - sqj;
            }
        }
        __syncthreads();
        if (lane < 16) {
            #pragma unroll 4
            for (int cc = 0; cc < 64; ++cc) {
                const float v = dist[lane][cc];
                const int j = n0 + cc;
                if (v > b0v) {
                    b2v = b1v; b2i = b1i;
                    b1v = b0v; b1i = b0i;
                    b0v = v;   b0i = j;
                } else if (v > b1v) {
                    b2v = b1v; b2i = b1i;
                    b1v = v;   b1i = j;
                } else if (v > b2v) {
                    b2v = v;   b2i = j;
                }
            }
        }
        __syncthreads();
    }

    if (lane < 16) {
        const int row = m0 + lane;
        idx[row * 3 + 0] = b0i;
        idx[row * 3 + 1] = b1i;
        idx[row * 3 + 2] = b2i;
    }
}

// ---------------------------------------------------------------------------
// SMOTE synthesis: new_fea[i*3+k] = a + w[i,k]*(b - a), a = x[idx[i,0]],
// b = x[idx[i,k]]. One block per output row, written straight into d_out tail.
// ---------------------------------------------------------------------------
__global__ __launch_bounds__(256) void synth_rows(
    const float* __restrict__ Xp, const int* __restrict__ idx,
    const float* __restrict__ lw, float* __restrict__ out)
{
    const int row = blockIdx.x;           // 0..NNEW-1
    const int i = row / 3, k = row % 3;
    const int ja = idx[i * 3 + 0];
    const int jb = idx[i * 3 + k];
    const float w = lw[i * 3 + k];
    const float* A = Xp + (size_t)ja * DD;
    const float* B = Xp + (size_t)jb * DD;
    float* o = out + (size_t)(BS + row) * DD;
    for (int d = threadIdx.x; d < DD; d += blockDim.x) {
        const float av = A[d];
        o[d] = av + w * (B[d] - av);
    }
}

__global__ void write_labels(const int* __restrict__ lbl, int* __restrict__ lo)
{
    const int i = blockIdx.x * blockDim.x + threadIdx.x;
    if (i < OUTROWS) lo[i] = (i < BS) ? lbl[i] : 1;
}

// ---------------------------------------------------------------------------
extern "C" void kernel_launch(void* const* d_in, const int* in_sizes, int n_in,
                              void* d_out, int out_size, void* d_ws, size_t ws_size,
                              hipStream_t stream)
{
    const float* fea  = (const float*)d_in[0];
    const int*   lbl  = (const int*)  d_in[1];
    const float* W1   = (const float*)d_in[2];
    const float* b1   = (const float*)d_in[3];
    const float* g1   = (const float*)d_in[4];
    const float* be1  = (const float*)d_in[5];
    const float* W2   = (const float*)d_in[6];
    const float* b2   = (const float*)d_in[7];
    const float* g2   = (const float*)d_in[8];
    const float* be2  = (const float*)d_in[9];
    const float* lw   = (const float*)d_in[10];
    (void)in_sizes; (void)n_in; (void)out_size; (void)ws_size;

    float* out = (float*)d_out;

    // workspace layout
    float* Y1     = (float*)d_ws;                       // BS*DD
    float* Y2     = Y1 + (size_t)BS * DD;               // BS*DD
    float* sum    = Y2 + (size_t)BS * DD;               // DD
    float* sumsq  = sum + DD;                           // DD
    float* mean   = sumsq + DD;                         // DD
    float* invstd = mean + DD;                          // DD
    float* sq     = invstd + DD;                        // PP
    int*   idx    = (int*)(sq + PP);                    // PP*3

    const dim3 gblk(128);
    const dim3 ggrid(BS / 128, DD / 64);                // 4 waves x 32 rows per block

    // ---- layer 1: GEMM + BN + swish (in place on Y1) ----
    gemm_bias_f32<<<ggrid, gblk, 0, stream>>>(fea, W1, b1, Y1, BS);
    zero_f32<<<(2 * DD + 255) / 256, 256, 0, stream>>>(sum, 2 * DD);
    col_stats_accum<<<128, 512, 0, stream>>>(Y1, sum, sumsq, BS);
    col_stats_final<<<2, 256, 0, stream>>>(sum, sumsq, mean, invstd, BS);
    bn_swish<<<4096, 256, 0, stream>>>(Y1, mean, invstd, g1, be1, Y1, BS);

    // ---- layer 2: GEMM + BN + swish (h2 written directly into d_out head) ----
    gemm_bias_f32<<<ggrid, gblk, 0, stream>>>(Y1, W2, b2, Y2, BS);
    zero_f32<<<(2 * DD + 255) / 256, 256, 0, stream>>>(sum, 2 * DD);
    col_stats_accum<<<128, 512, 0, stream>>>(Y2, sum, sumsq, BS);
    col_stats_final<<<2, 256, 0, stream>>>(sum, sumsq, mean, invstd, BS);
    bn_swish<<<4096, 256, 0, stream>>>(Y2, mean, invstd, g2, be2, out, BS);

    // ---- 3-NN over the positive block (rows 0..PP-1 of h2) ----
    row_sq<<<PP / 8, 256, 0, stream>>>(out, sq);
    knn_top3<<<PP / 16, 32, 0, stream>>>(out, sq, idx);

    // ---- synthesize new rows + labels ----
    synth_rows<<<NNEW, 256, 0, stream>>>(out, idx, lw, out);
    write_labels<<<(OUTROWS + 255) / 256, 256, 0, stream>>>(
        lbl, (int*)(out + (size_t)OUTROWS * DD));
}